// SAGE_64141041599037
// MI455X (gfx1250) — compile-verified
//
#include <hip/hip_runtime.h>

#define N_NODES 100000
#define N_EDGES 1600000
#define IN_F 128
#define HID_F 256
#define OUT_F 64

typedef __attribute__((ext_vector_type(2))) float v2f;
typedef __attribute__((ext_vector_type(8))) float v8f;

// ---------------------------------------------------------------------------
// zero-init workspace regions (harness poisons ws with 0xAA before timing)
// ---------------------------------------------------------------------------
__global__ void zero_f32(float* __restrict__ p, int n) {
    int i = blockIdx.x * blockDim.x + threadIdx.x;
    int stride = gridDim.x * blockDim.x;
    for (; i < n; i += stride) p[i] = 0.0f;
}

// ---------------------------------------------------------------------------
// Layer-1 scatter: agg1[dst] += x[src]  (128 floats/edge, 32 lanes x float4)
// lane 0 of each edge also accumulates the in-degree.
// ---------------------------------------------------------------------------
__global__ void scatter1(const float* __restrict__ x,
                         const int* __restrict__ src,
                         const int* __restrict__ dst,
                         float* __restrict__ agg1,
                         float* __restrict__ deg) {
    unsigned t = blockIdx.x * blockDim.x + threadIdx.x;
    unsigned e = t >> 5;
    if (e >= (unsigned)N_EDGES) return;
    unsigned c = (t & 31u) * 4u;
    int s = src[e];
    int d = dst[e];
    const float4 v = *(const float4*)(x + (size_t)s * IN_F + c);
    float* o = agg1 + (size_t)d * IN_F + c;
    atomicAdd(o + 0, v.x);
    atomicAdd(o + 1, v.y);
    atomicAdd(o + 2, v.z);
    atomicAdd(o + 3, v.w);
    if ((t & 31u) == 0u) atomicAdd(deg + d, 1.0f);
}

// ---------------------------------------------------------------------------
// Layer-2 scatter: agg2[dst] += h[src]  (256 floats/edge, 64 lanes x float4)
// ---------------------------------------------------------------------------
__global__ void scatter2(const float* __restrict__ h,
                         const int* __restrict__ src,
                         const int* __restrict__ dst,
                         float* __restrict__ agg2) {
    unsigned t = blockIdx.x * blockDim.x + threadIdx.x;
    unsigned e = t >> 6;
    if (e >= (unsigned)N_EDGES) return;
    unsigned c = (t & 63u) * 4u;
    int s = src[e];
    int d = dst[e];
    const float4 v = *(const float4*)(h + (size_t)s * HID_F + c);
    float* o = agg2 + (size_t)d * HID_F + c;
    atomicAdd(o + 0, v.x);
    atomicAdd(o + 1, v.y);
    atomicAdd(o + 2, v.z);
    atomicAdd(o + 3, v.w);
}

// ---------------------------------------------------------------------------
// GEMM 1: h = relu( ((agg1 + x) / (deg+1)) @ W1n + b1 )   [N x 128] @ [128 x 256]
// One wave per 16x16 output tile, v_wmma_f32_16x16x4_f32 over K=128.
// ---------------------------------------------------------------------------
__global__ void __launch_bounds__(128) gemm1(const float* __restrict__ x,
                                             const float* __restrict__ agg1,
                                             const float* __restrict__ deg,
                                             const float* __restrict__ W1n,
                                             const float* __restrict__ b1,
                                             float* __restrict__ h) {
    int wave = threadIdx.x >> 5;
    int lane = threadIdx.x & 31;
    int tile = blockIdx.x * 4 + wave;      // 6250 * 16 tiles total
    int rowBase = (tile >> 4) * 16;
    int colBase = (tile & 15) * 16;
    int g  = lane >> 4;                    // lane group (0 or 1)
    int nl = lane & 15;
    int m = rowBase + nl;                  // A row handled by this lane
    int n = colBase + nl;                  // B/C/D column handled by this lane

    float r = 1.0f / (deg[m] + 1.0f);
    const float* arow = agg1 + (size_t)m * IN_F;
    const float* xrow = x    + (size_t)m * IN_F;

    v8f acc = {};
    for (int kk = 0; kk < IN_F; kk += 4) {
        int kv = kk + 2 * g;               // A/B VGPR-v holds K = v + 2g
        v2f ta = *(const v2f*)(arow + kv);
        v2f tx = *(const v2f*)(xrow + kv);
        v2f a;
        a.x = (ta.x + tx.x) * r;
        a.y = (ta.y + tx.y) * r;
        v2f b;
        b.x = W1n[(size_t)kv * HID_F + n];
        b.y = W1n[(size_t)(kv + 1) * HID_F + n];
        acc = __builtin_amdgcn_wmma_f32_16x16x4_f32(false, a, false, b,
                                                    (short)0, acc, false, false);
    }

    float bias = b1[n];
#pragma unroll
    for (int v = 0; v < 8; ++v) {          // C/D: element (m = v + 8g, n)
        int row = rowBase + v + 8 * g;
        float val = acc[v] + bias;
        h[(size_t)row * HID_F + n] = val > 0.0f ? val : 0.0f;
    }
}

// ---------------------------------------------------------------------------
// GEMM 2: out = h @ W2s + (agg2 / max(deg,1)) @ W2n + b2   [N x 256] @ [256 x 64]
// Two WMMAs per K-step sharing one accumulator.
// ---------------------------------------------------------------------------
__global__ void __launch_bounds__(128) gemm2(const float* __restrict__ h,
                                             const float* __restrict__ agg2,
                                             const float* __restrict__ deg,
                                             const float* __restrict__ W2s,
                                             const float* __restrict__ W2n,
                                             const float* __restrict__ b2,
                                             float* __restrict__ out) {
    int wave = threadIdx.x >> 5;
    int lane = threadIdx.x & 31;
    int tile = blockIdx.x * 4 + wave;      // 6250 * 4 tiles total
    int rowBase = (tile >> 2) * 16;
    int colBase = (tile & 3) * 16;
    int g  = lane >> 4;
    int nl = lane & 15;
    int m = rowBase + nl;
    int n = colBase + nl;

    float dm = deg[m];
    float r = 1.0f / (dm > 1.0f ? dm : 1.0f);
    const float* hrow = h    + (size_t)m * HID_F;
    const float* arow = agg2 + (size_t)m * HID_F;

    v8f acc = {};
    for (int kk = 0; kk < HID_F; kk += 4) {
        int kv = kk + 2 * g;
        // self term: h @ W2s
        v2f a1 = *(const v2f*)(hrow + kv);
        v2f bs;
        bs.x = W2s[(size_t)kv * OUT_F + n];
        bs.y = W2s[(size_t)(kv + 1) * OUT_F + n];
        acc = __builtin_amdgcn_wmma_f32_16x16x4_f32(false, a1, false, bs,
                                                    (short)0, acc, false, false);
        // neighbor term: (agg2/deg) @ W2n
        v2f ta = *(const v2f*)(arow + kv);
        v2f a2;
        a2.x = ta.x * r;
        a2.y = ta.y * r;
        v2f bn;
        bn.x = W2n[(size_t)kv * OUT_F + n];
        bn.y = W2n[(size_t)(kv + 1) * OUT_F + n];
        acc = __builtin_amdgcn_wmma_f32_16x16x4_f32(false, a2, false, bn,
                                                    (short)0, acc, false, false);
    }

    float bias = b2[n];
#pragma unroll
    for (int v = 0; v < 8; ++v) {
        int row = rowBase + v + 8 * g;
        out[(size_t)row * OUT_F + n] = acc[v] + bias;
    }
}

// ---------------------------------------------------------------------------
// launch
// ---------------------------------------------------------------------------
extern "C" void kernel_launch(void* const* d_in, const int* in_sizes, int n_in,
                              void* d_out, int out_size, void* d_ws, size_t ws_size,
                              hipStream_t stream) {
    const float* x   = (const float*)d_in[0];
    const int*   src = (const int*)d_in[1];
    const int*   dst = (const int*)d_in[2];
    const float* W1n = (const float*)d_in[3];
    const float* b1  = (const float*)d_in[4];
    const float* W2s = (const float*)d_in[5];
    const float* W2n = (const float*)d_in[6];
    const float* b2  = (const float*)d_in[7];
    float* out = (float*)d_out;
    float* ws  = (float*)d_ws;

    // workspace layout (floats): deg | pad | agg1 | h | agg2  (~256.4 MB)
    const size_t DEG_OFF  = 0;
    const size_t AGG1_OFF = 102400;                              // deg + pad
    const size_t H_OFF    = AGG1_OFF + (size_t)N_NODES * IN_F;   // + 12.8M
    const size_t AGG2_OFF = H_OFF    + (size_t)N_NODES * HID_F;  // + 25.6M
    float* deg  = ws + DEG_OFF;
    float* agg1 = ws + AGG1_OFF;
    float* h    = ws + H_OFF;
    float* agg2 = ws + AGG2_OFF;

    // 1) zero deg + agg1 region
    int nz1 = (int)(AGG1_OFF + (size_t)N_NODES * IN_F);
    zero_f32<<<4096, 256, 0, stream>>>(ws, nz1);

    // 2) layer-1 scatter (32 lanes/edge): 1.6M*32/256 = 200000 blocks
    scatter1<<<(N_EDGES * 32u) / 256u, 256, 0, stream>>>(x, src, dst, agg1, deg);

    // 3) GEMM1 + bias + relu -> h : 100000 tiles / 4 waves per block
    gemm1<<<(N_NODES / 16) * (HID_F / 16) / 4, 128, 0, stream>>>(x, agg1, deg, W1n, b1, h);

    // 4) zero agg2
    zero_f32<<<4096, 256, 0, stream>>>(agg2, N_NODES * HID_F);

    // 5) layer-2 scatter (64 lanes/edge): 1.6M*64/256 = 400000 blocks
    scatter2<<<(N_EDGES * 64u) / 256u, 256, 0, stream>>>(h, src, dst, agg2);

    // 6) GEMM2 (self + neighbor) + bias -> out : 25000 tiles / 4 per block
    gemm2<<<(N_NODES / 16) * (OUT_F / 16) / 4, 128, 0, stream>>>(h, agg2, deg, W2s, W2n, b2, out);
}